// VideoFlickeringAttack_74809740361781
// MI455X (gfx1250) — compile-verified
//
#include <hip/hip_runtime.h>
#include <hip/hip_bf16.h>
#include <math.h>

typedef __attribute__((ext_vector_type(2))) float v2f;
typedef __attribute__((ext_vector_type(8))) float v8f;

#define NB      8
#define FRAMES  256                 // S*T
#define CCH     3
#define HH      224
#define WW      224
#define PIX     (CCH*HH*WW)         // 150528
#define BS      56
#define EPSV    0.1f
#define UP_ELEMS (PIX*NB)           // 1204224

// ---------------- Stage 1: tanh(coeffs) transposed + bilinear upsample transposed ----
// tanhc[f*8 + b] = tanh(coeffs[b,f])      (256 x 8, f-major)
// upT[p*8 + b]   = bilinear(basis[b], p)  (150528 x 8, pixel-major)
__global__ void vfa_prep(const float* __restrict__ basis,
                         const float* __restrict__ coeffs,
                         float* __restrict__ tanhc,
                         float* __restrict__ upT) {
  int tid = blockIdx.x * blockDim.x + threadIdx.x;
  if (tid < FRAMES * NB) {
    int f = tid >> 3, b = tid & 7;
    tanhc[tid] = tanhf(coeffs[b * FRAMES + f]);
  }
  if (tid >= UP_ELEMS) return;
  int b = tid & 7;
  int p = tid >> 3;
  int c = p / (HH * WW);
  int r = p - c * (HH * WW);
  int h = r / WW;
  int w = r - h * WW;
  // half-pixel centers, scale 4x:  src = (x+0.5)/4 - 0.5
  float sh = 0.25f * (float)h - 0.375f;
  float sw = 0.25f * (float)w - 0.375f;
  float fh = floorf(sh), fw = floorf(sw);
  float th = sh - fh, tw = sw - fw;
  int ih = (int)fh, iw = (int)fw;
  int h0 = min(max(ih, 0), BS - 1), h1 = min(max(ih + 1, 0), BS - 1);
  int w0 = min(max(iw, 0), BS - 1), w1 = min(max(iw + 1, 0), BS - 1);
  const float* bp = basis + ((size_t)b * CCH + c) * (BS * BS);
  float v00 = bp[h0 * BS + w0], v01 = bp[h0 * BS + w1];
  float v10 = bp[h1 * BS + w0], v11 = bp[h1 * BS + w1];
  float t0 = v00 + tw * (v01 - v00);
  float t1 = v10 + tw * (v11 - v10);
  upT[tid] = t0 + th * (t1 - t0);
}

// ---------------- Stage 2: Out[256 x 150528] = tanhc(256x8) * upT^T(8x150528), clip ----
// One wave = one 16-frame block; loops over 16 contiguous 16-pixel tiles.
// Per tile: two chained V_WMMA_F32_16X16X4_F32 (K = 0..3, then 4..7), exact f32.
#define FB       (FRAMES / 16)      // 16 frame blocks
#define PT       (PIX / 16)         // 9408 pixel tiles
#define TPW      16                 // pixel tiles per wave
#define WAVES_P  (PT / TPW)         // 588
#define NBLOCKS  (FB * WAVES_P / 8) // 1176 blocks of 8 waves

__global__ void __launch_bounds__(256) vfa_wmma(const float* __restrict__ tanhc,
                                                const float* __restrict__ upT,
                                                float* __restrict__ out) {
  const int wave   = blockIdx.x * (blockDim.x >> 5) + (threadIdx.x >> 5);
  const int lane   = threadIdx.x & 31;
  const int fBlock = wave & (FB - 1);     // FB == 16
  const int pw     = wave >> 4;           // 0 .. WAVES_P-1
  const int M      = lane & 15;           // A row / B col / D col
  const int K0     = (lane >> 4) << 1;    // K split: lanes 0-15 -> K0=0, lanes 16-31 -> K0=2
  const int f      = fBlock * 16 + M;

  // A fragment (coeffs), reused for all 16 pixel tiles of this wave.
  const v2f aLo = *(const v2f*)(tanhc + f * NB + K0);       // K = K0, K0+1
  const v2f aHi = *(const v2f*)(tanhc + f * NB + K0 + 4);   // K = K0+4, K0+5

  const int rowHalf = (lane >> 4) << 3;   // D: VGPR r holds M = r + rowHalf
  float* outBase = out + (size_t)fBlock * 16 * PIX;

#pragma unroll 1
  for (int t = 0; t < TPW; ++t) {
    const int pix = (pw * TPW + t) * 16 + M;                // N = lane&15
    const float* bp = upT + (size_t)pix * NB;
    v2f bLo = *(const v2f*)(bp + K0);
    v2f bHi = *(const v2f*)(bp + K0 + 4);

    v8f acc = {};
    acc = __builtin_amdgcn_wmma_f32_16x16x4_f32(false, aLo, false, bLo,
                                                (short)0, acc, false, false);
    acc = __builtin_amdgcn_wmma_f32_16x16x4_f32(false, aHi, false, bHi,
                                                (short)0, acc, false, false);
#pragma unroll
    for (int rr = 0; rr < 8; ++rr) {
      float v = fminf(fmaxf(acc[rr], -EPSV), EPSV);
      // Output is write-once streaming: bypass cache with non-temporal store.
      __builtin_nontemporal_store(v, outBase + (size_t)(rr + rowHalf) * PIX + pix);
    }
  }
}

extern "C" void kernel_launch(void* const* d_in, const int* in_sizes, int n_in,
                              void* d_out, int out_size, void* d_ws, size_t ws_size,
                              hipStream_t stream) {
  (void)in_sizes; (void)n_in; (void)out_size; (void)ws_size;
  const float* basis  = (const float*)d_in[0];   // (8,3,56,56)
  const float* coeffs = (const float*)d_in[1];   // (8,256)
  float* tanhc = (float*)d_ws;                   // 2048 floats
  float* upT   = tanhc + FRAMES * NB;            // 1204224 floats (~4.6 MiB)
  float* out   = (float*)d_out;                  // (1,16,16,3,224,224) f32

  const int prepBlocks = (UP_ELEMS + 255) / 256;
  vfa_prep<<<prepBlocks, 256, 0, stream>>>(basis, coeffs, tanhc, upT);
  vfa_wmma<<<NBLOCKS, 256, 0, stream>>>(tanhc, upT, out);
}